// SPADE_27204322853709
// MI455X (gfx1250) — compile-verified
//
#include <hip/hip_runtime.h>
#include <hip/hip_bf16.h>
#include <math.h>

typedef float v2f __attribute__((ext_vector_type(2)));
typedef float v8f __attribute__((ext_vector_type(8)));

#define IMG 224
#define NLIB 128
#define ZDIM 1024

// ---------------------------------------------------------------------------
// Kernel 1: dist2[n] = sum_c (z_lib[n,c]-z[c])^2 for n=0..127 via WMMA.
// Gram-diagonal trick: D = diff * diff^T with V_WMMA_F32_16X16X4_F32; the
// diagonal of D accumulates the squared L2 distance exactly in fp32.
// One wave per 16-row tile; 8 waves total (256 threads, one block).
// A layout (ISA 7.12.2, 32-bit A 16x4): lane l, VGPR j -> M=l%16, K=j+2*(l/16).
// B assumed mirrored (same per-lane index function), so the same registers
// serve as both operands.
// ---------------------------------------------------------------------------
__global__ void __launch_bounds__(256)
spade_dist_wmma(const float* __restrict__ z,
                const float* __restrict__ z_lib,
                float* __restrict__ dist2) {
  const int wave    = threadIdx.x >> 5;       // 0..7
  const int lane    = threadIdx.x & 31;
  const int rowbase = wave * 16;
  const int m       = lane & 15;
  const int khalf   = (lane >> 4) << 1;       // 0 or 2
  const float* rowp = z_lib + (size_t)(rowbase + m) * ZDIM;

  v8f acc = {};
  #pragma unroll 4
  for (int c = 0; c < ZDIM; c += 4) {
    const int kk = c + khalf;
    v2f d;
    d.x = rowp[kk]     - z[kk];
    d.y = rowp[kk + 1] - z[kk + 1];
    // D = A*B + C with A = B = diff  =>  Gram matrix; diag = partial dist^2
    acc = __builtin_amdgcn_wmma_f32_16x16x4_f32(
        /*neg_a=*/false, d, /*neg_b=*/false, d,
        /*c_mod=*/(short)0, acc, /*reuse_a=*/false, /*reuse_b=*/false);
  }

  // Diagonal extraction per C/D layout: D[v,v] at (VGPR v, lane v);
  // D[v+8,v+8] at (VGPR v, lane v+24).
  #pragma unroll
  for (int v = 0; v < 8; ++v) {
    float lo = __shfl(acc[v], v, 32);
    float hi = __shfl(acc[v], v + 24, 32);
    if (lane == 0) {
      dist2[rowbase + v]     = lo;
      dist2[rowbase + 8 + v] = hi;
    }
  }
}

// ---------------------------------------------------------------------------
// Kernel 2: stable top-k (k smallest dist) by rank counting; writes z_score
// (mean of sqrt(dist2) over selected) and the compact selected-index list.
// ---------------------------------------------------------------------------
__global__ void __launch_bounds__(128)
spade_topk(const float* __restrict__ dist2,
           const int* __restrict__ kp,
           float* __restrict__ zscore_out,
           int* __restrict__ sel) {
  __shared__ float d[NLIB];
  __shared__ float s[NLIB];
  const int i = threadIdx.x;
  const int k = kp[0];
  const float di = dist2[i];
  d[i] = di;
  __syncthreads();

  int rank = 0;
  #pragma unroll 8
  for (int j = 0; j < NLIB; ++j) {
    const float dj = d[j];
    rank += (dj < di || (dj == di && j < i)) ? 1 : 0;
  }
  if (rank < k) sel[rank] = i;

  s[i] = (rank < k) ? sqrtf(di) : 0.0f;
  __syncthreads();
  for (int stride = NLIB / 2; stride > 0; stride >>= 1) {
    if (i < stride) s[i] += s[i + stride];
    __syncthreads();
  }
  if (i == 0) zscore_out[0] = s[0] / (float)k;
}

// ---------------------------------------------------------------------------
// Kernel 3: initialize min buffers to +inf (bit pattern, for uint atomicMin)
// ---------------------------------------------------------------------------
__global__ void spade_init_inf(unsigned int* __restrict__ p, int n) {
  int i = blockIdx.x * blockDim.x + threadIdx.x;
  if (i < n) p[i] = 0x7F800000u;  // +inf
}

// ---------------------------------------------------------------------------
// Kernel 4: per-level min of squared channel distance over selected neighbors.
// grid.y indexes neighbor slot n (guard n<k). The 280MB lib stream is staged
// through LDS with GLOBAL_LOAD_ASYNC_TO_LDS_B128 (CDNA5 async path, ASYNCcnt):
// double-buffered chunks of 8 channels, 16 async b128 in flight per thread,
// zero VGPR staging cost. Each lane consumes exactly the bytes it fetched, so
// only s_wait_asynccnt (per-wave) is needed -- no workgroup barrier.
// fmap (3.2MB, L2-resident) stays on direct loads. Result folded into global
// atomicMin on uint bit patterns (valid: values are nonnegative floats).
// ---------------------------------------------------------------------------
#define LVL_TPB 128
#define LVL_CH  8

__global__ void __launch_bounds__(LVL_TPB)
spade_level_min(const float* __restrict__ lib,
                const float* __restrict__ fmap,
                const int* __restrict__ sel,
                const int* __restrict__ kp,
                unsigned int* __restrict__ sminsq_bits,
                int C, int P) {
  __shared__ float4 buf[2][LVL_CH][LVL_TPB];

  const int n = blockIdx.y;
  const int np4 = P >> 2;
  const int p4 = blockIdx.x * LVL_TPB + threadIdx.x;
  // No barriers below: safe early-out for idle threads.
  if (n >= kp[0] || p4 >= np4) return;

  const int tid = threadIdx.x;
  const int idx = sel[n];
  const int NC = C / LVL_CH;  // 32 / 64 / 128 (always even)
  const float4* __restrict__ fm = (const float4*)fmap;
  const float* gbase = lib + (size_t)idx * C * P + (size_t)(p4 << 2);

  // Issue one chunk (LVL_CH channels) of async global->LDS b128 loads.
  auto issue = [&](int bsel, int ci) {
    const float* g = gbase + (size_t)ci * LVL_CH * P;
    #pragma unroll
    for (int ch = 0; ch < LVL_CH; ++ch) {
      // Flat-pointer low 32 bits == LDS byte offset (aperture mapping).
      unsigned int lds_addr =
          (unsigned int)(uintptr_t)&buf[bsel][ch][tid];
      unsigned long long gaddr =
          (unsigned long long)(uintptr_t)(g + (size_t)ch * P);
      asm volatile("global_load_async_to_lds_b128 %0, %1, off"
                   :: "v"(lds_addr), "v"(gaddr)
                   : "memory");
    }
  };

  issue(0, 0);
  if (NC > 1) issue(1, 1);

  float4 acc = make_float4(0.f, 0.f, 0.f, 0.f);
  for (int ci = 0; ci < NC; ++ci) {
    if (ci + 1 < NC) {
      // Two chunks (16 ops) outstanding: <=8 means chunk ci has landed.
      asm volatile("s_wait_asynccnt 0x8" ::: "memory");
    } else {
      asm volatile("s_wait_asynccnt 0x0" ::: "memory");
    }
    const int b = ci & 1;
    const int c0 = ci * LVL_CH;
    #pragma unroll
    for (int ch = 0; ch < LVL_CH; ++ch) {
      const float4 l = buf[b][ch][tid];
      const float4 f = fm[(size_t)(c0 + ch) * np4 + p4];
      const float dx = l.x - f.x, dy = l.y - f.y;
      const float dz = l.z - f.z, dw = l.w - f.w;
      acc.x += dx * dx; acc.y += dy * dy;
      acc.z += dz * dz; acc.w += dw * dw;
    }
    if (ci + 2 < NC) issue(b, ci + 2);  // buffer b fully consumed above
  }

  unsigned int* o = sminsq_bits + 4 * p4;
  atomicMin(o + 0, __float_as_uint(acc.x));
  atomicMin(o + 1, __float_as_uint(acc.y));
  atomicMin(o + 2, __float_as_uint(acc.z));
  atomicMin(o + 3, __float_as_uint(acc.w));
}

// ---------------------------------------------------------------------------
// Kernel 5: bilinear upsample (half-pixel centers) of sqrt(min-sq) per level,
// summed into s_map [224x224].
// ---------------------------------------------------------------------------
__device__ __forceinline__ float bil_sqrt(const float* __restrict__ s, int S,
                                          int ox, int oy) {
  const float scale = (float)S / (float)IMG;
  const float sx = ((float)ox + 0.5f) * scale - 0.5f;
  const float sy = ((float)oy + 0.5f) * scale - 0.5f;
  int x0 = (int)floorf(sx); const float fx = sx - (float)x0;
  int y0 = (int)floorf(sy); const float fy = sy - (float)y0;
  int x1 = x0 + 1, y1 = y0 + 1;
  x0 = min(max(x0, 0), S - 1); x1 = min(max(x1, 0), S - 1);
  y0 = min(max(y0, 0), S - 1); y1 = min(max(y1, 0), S - 1);
  const float v00 = sqrtf(s[y0 * S + x0]);
  const float v01 = sqrtf(s[y0 * S + x1]);
  const float v10 = sqrtf(s[y1 * S + x0]);
  const float v11 = sqrtf(s[y1 * S + x1]);
  return (1.f - fy) * ((1.f - fx) * v00 + fx * v01) +
         fy         * ((1.f - fx) * v10 + fx * v11);
}

__global__ void __launch_bounds__(256)
spade_upsum(const float* __restrict__ s0, const float* __restrict__ s1,
            const float* __restrict__ s2, float* __restrict__ smap) {
  const int i = blockIdx.x * blockDim.x + threadIdx.x;
  if (i >= IMG * IMG) return;
  const int y = i / IMG, x = i % IMG;
  smap[i] = bil_sqrt(s0, 56, x, y) + bil_sqrt(s1, 28, x, y) +
            bil_sqrt(s2, 14, x, y);
}

// ---------------------------------------------------------------------------
// Kernels 6/7: separable 17-tap Gaussian (sigma=4), reflect boundary.
// Weights constant-fold (expf of literals in an unrolled loop).
// ---------------------------------------------------------------------------
__device__ __forceinline__ void gauss17(float* w) {
  float s = 0.f;
  #pragma unroll
  for (int t = 0; t < 17; ++t) {
    const float x = (float)(t - 8) * 0.25f;
    w[t] = expf(-0.5f * x * x);
    s += w[t];
  }
  const float inv = 1.0f / s;
  #pragma unroll
  for (int t = 0; t < 17; ++t) w[t] *= inv;
}

__device__ __forceinline__ int reflect224(int i) {
  if (i < 0) i = -i;
  if (i > IMG - 1) i = 2 * (IMG - 1) - i;
  return i;
}

__global__ void __launch_bounds__(256)
spade_blur_h(const float* __restrict__ in, float* __restrict__ out) {
  const int i = blockIdx.x * blockDim.x + threadIdx.x;
  if (i >= IMG * IMG) return;
  const int y = i / IMG, x = i % IMG;
  float w[17]; gauss17(w);
  float acc = 0.f;
  #pragma unroll
  for (int t = 0; t < 17; ++t) acc += w[t] * in[y * IMG + reflect224(x + t - 8)];
  out[i] = acc;
}

__global__ void __launch_bounds__(256)
spade_blur_v(const float* __restrict__ in, float* __restrict__ out) {
  const int i = blockIdx.x * blockDim.x + threadIdx.x;
  if (i >= IMG * IMG) return;
  const int y = i / IMG, x = i % IMG;
  float w[17]; gauss17(w);
  float acc = 0.f;
  #pragma unroll
  for (int t = 0; t < 17; ++t) acc += w[t] * in[reflect224(y + t - 8) * IMG + x];
  out[i] = acc;
}

// ---------------------------------------------------------------------------
extern "C" void kernel_launch(void* const* d_in, const int* in_sizes, int n_in,
                              void* d_out, int out_size, void* d_ws, size_t ws_size,
                              hipStream_t stream) {
  const float* z     = (const float*)d_in[0];
  const float* z_lib = (const float*)d_in[1];
  const float* fmap0 = (const float*)d_in[2];
  const float* fmap1 = (const float*)d_in[3];
  const float* fmap2 = (const float*)d_in[4];
  const float* lib0  = (const float*)d_in[5];
  const float* lib1  = (const float*)d_in[6];
  const float* lib2  = (const float*)d_in[7];
  const int*   kp    = (const int*)d_in[8];
  float* out = (float*)d_out;   // out[0]=z_score, out[1..50176]=blurred map

  // Workspace layout (floats)
  float* ws     = (float*)d_ws;
  float* dist2  = ws;                    // 128
  int*   sel    = (int*)(ws + 128);      // 128 ints
  float* s0     = ws + 256;              // 56*56 = 3136 (min of squared dist)
  float* s1     = s0 + 3136;             // 28*28 = 784
  float* s2     = s1 + 784;              // 14*14 = 196
  float* smap   = s2 + 196;              // 224*224
  float* tmp    = smap + IMG * IMG;      // 224*224

  // 1) image-level squared distances via WMMA Gram diagonal
  spade_dist_wmma<<<1, 256, 0, stream>>>(z, z_lib, dist2);

  // 2) top-k (k smallest), z_score, compact index list
  spade_topk<<<1, 128, 0, stream>>>(dist2, kp, out, sel);

  // 3) init min buffers to +inf (3136+784+196 = 4116 floats)
  spade_init_inf<<<(4116 + 255) / 256, 256, 0, stream>>>((unsigned int*)s0, 4116);

  // 4) per-level async-LDS streaming min over selected neighbors
  {
    dim3 g0((784 + LVL_TPB - 1) / LVL_TPB, NLIB);   // level0: 784 float4 lanes
    spade_level_min<<<g0, LVL_TPB, 0, stream>>>(lib0, fmap0, sel, kp,
                                                (unsigned int*)s0, 256, 3136);
    dim3 g1((196 + LVL_TPB - 1) / LVL_TPB, NLIB);   // level1: 196 float4 lanes
    spade_level_min<<<g1, LVL_TPB, 0, stream>>>(lib1, fmap1, sel, kp,
                                                (unsigned int*)s1, 512, 784);
    dim3 g2(1, NLIB);                                // level2: 49 float4 lanes
    spade_level_min<<<g2, LVL_TPB, 0, stream>>>(lib2, fmap2, sel, kp,
                                                (unsigned int*)s2, 1024, 196);
  }

  // 5) bilinear upsample of sqrt + sum across levels
  spade_upsum<<<(IMG * IMG + 255) / 256, 256, 0, stream>>>(s0, s1, s2, smap);

  // 6) separable Gaussian blur (reflect), final into out+1
  spade_blur_h<<<(IMG * IMG + 255) / 256, 256, 0, stream>>>(smap, tmp);
  spade_blur_v<<<(IMG * IMG + 255) / 256, 256, 0, stream>>>(tmp, out + 1);
}